// e2e_MLP_75316546503013
// MI455X (gfx1250) — compile-verified
//
#include <hip/hip_runtime.h>

// ---------------------------------------------------------------------------
// e2e MLP (4->30->30->1, ReLU) + merit-order economic dispatch, fused.
// MI455X / gfx1250: wave32, V_WMMA_F32_16X16X4_F32 for all matmuls (fp32
// exact), LDS bounce for the C-layout -> A-layout transpose, coalesced
// float4 stores for the 104.8 MB output (the real bottleneck: ~4.5us @ 23.3TB/s).
// ---------------------------------------------------------------------------

typedef __attribute__((ext_vector_type(2))) float v2f;
typedef __attribute__((ext_vector_type(4))) float v4f;
typedef __attribute__((ext_vector_type(8))) float v8f;

#define NU    100   // number of units
#define HDIM  30    // hidden width
#define INDIM 4
#define TILES_PER_WAVE 4
#define WAVES_PER_BLOCK 8
#define H1_PITCH 34  // LDS row pitch in floats: conflict-free for stride-128B rows

// ---------------------------------------------------------------------------
// Setup: per-unit exclusive cumulative capacity in ORIGINAL unit order
// (cum[u] = sum of Pmax[j] over units j strictly cheaper, stable tie-break
// by index -> matches jnp.argsort + exclusive cumsum + scatter), plus sum(Pd).
// Runs once, 1 block, O(100^2) — negligible.
// ---------------------------------------------------------------------------
__global__ void setup_units_kernel(const float* __restrict__ Cost,
                                   const float* __restrict__ Pmax,
                                   const float* __restrict__ Pd,
                                   float* __restrict__ cum,
                                   float* __restrict__ pmx,
                                   float* __restrict__ sumPd) {
    int t = threadIdx.x;
    if (t < NU) {
        float ci = Cost[t];
        float acc = 0.0f;
        for (int j = 0; j < NU; ++j) {
            float cj = Cost[j];
            bool before = (cj < ci) || ((cj == ci) && (j < t));
            acc += before ? Pmax[j] : 0.0f;
        }
        cum[t] = acc;
        pmx[t] = Pmax[t];
    }
    if (t == 0) {
        float s = 0.0f;
        for (int j = 0; j < NU; ++j) s += Pd[j];
        *sumPd = s;
    }
}

// ---------------------------------------------------------------------------
// Main fused kernel. Each wave handles TILES_PER_WAVE tiles of 16 samples.
// ---------------------------------------------------------------------------
__launch_bounds__(WAVES_PER_BLOCK * 32)
__global__ void mlp_dispatch_kernel(const float* __restrict__ x,
                                    const float* __restrict__ W1,
                                    const float* __restrict__ b1,
                                    const float* __restrict__ W2,
                                    const float* __restrict__ b2,
                                    const float* __restrict__ W3,
                                    const float* __restrict__ b3,
                                    const float* __restrict__ wcap,
                                    const float* __restrict__ cum,
                                    const float* __restrict__ pmx,
                                    const float* __restrict__ sumPd,
                                    float* __restrict__ out,
                                    int Btotal) {
    __shared__ float s_cum[NU];
    __shared__ float s_pmx[NU];
    __shared__ __align__(16) float s_h1[WAVES_PER_BLOCK][16 * H1_PITCH];
    __shared__ float s_td[WAVES_PER_BLOCK][16];

    const int tid    = threadIdx.x;
    const int wave   = tid >> 5;
    const int lane   = tid & 31;
    const int lane15 = lane & 15;
    const int hi     = lane >> 4;        // 0: lanes 0-15, 1: lanes 16-31

    // Cooperative load of the per-unit dispatch tables.
    if (tid < NU) { s_cum[tid] = cum[tid]; s_pmx[tid] = pmx[tid]; }
    __syncthreads();

    // --- Per-lane weight operands in WMMA f32 16x16x4 B-matrix layout ------
    // B (4x16, KxN): lane holds column n=lane15; hi=0 -> rows {0,1} in
    // (.x,.y); hi=1 -> rows {2,3}. N padded 30->32 with zeros, K padded
    // 30->32 with zeros (layer 2) so pad columns compute to relu(0)=0.
    v2f w1op[2];
    float bias1[2], bias2[2];
    #pragma unroll
    for (int t = 0; t < 2; ++t) {
        int col = t * 16 + lane15;
        int kb  = 2 * hi;
        bool cv = (col < HDIM);
        w1op[t].x = cv ? W1[kb * HDIM + col]       : 0.0f;
        w1op[t].y = cv ? W1[(kb + 1) * HDIM + col] : 0.0f;
        bias1[t]  = cv ? b1[col] : 0.0f;
        bias2[t]  = cv ? b2[col] : 0.0f;
    }
    v2f w2op[8][2];
    #pragma unroll
    for (int c = 0; c < 8; ++c) {
        #pragma unroll
        for (int t = 0; t < 2; ++t) {
            int kb  = 4 * c + 2 * hi;
            int col = t * 16 + lane15;
            bool cv = (col < HDIM);
            w2op[c][t].x = (cv && (kb     < HDIM)) ? W2[kb * HDIM + col]       : 0.0f;
            w2op[c][t].y = (cv && (kb + 1 < HDIM)) ? W2[(kb + 1) * HDIM + col] : 0.0f;
        }
    }
    const float w3a   = W3[lane15];                                    // cols 0-15
    const float w3b   = (16 + lane15 < HDIM) ? W3[16 + lane15] : 0.0f; // cols 16-29
    const float bias3 = b3[0];
    const float wc    = wcap[0];
    const float sPd   = sumPd[0];

    float* h1 = s_h1[wave];
    float* td = s_td[wave];
    const int waveBase = (blockIdx.x * WAVES_PER_BLOCK + wave) * (16 * TILES_PER_WAVE);

    for (int tile = 0; tile < TILES_PER_WAVE; ++tile) {
        const int b0 = waveBase + tile * 16;
        if (b0 + 16 > Btotal) break;   // wave-uniform; EXEC stays all-ones

        // ---- A operand: X tile (16x4). A layout: lane holds row m=lane15,
        //      hi=0 -> K {0,1}, hi=1 -> K {2,3}. One float2 per lane.
        const v2f a1 = *(const v2f*)(x + (size_t)(b0 + lane15) * INDIM + 2 * hi);

        // ---- Layer 1: [16x4] @ [4x32] -> two 16x16 WMMAs ----
        v8f c0 = {};
        v8f c1 = {};
        c0 = __builtin_amdgcn_wmma_f32_16x16x4_f32(false, a1, false, w1op[0],
                                                   (short)0, c0, false, false);
        c1 = __builtin_amdgcn_wmma_f32_16x16x4_f32(false, a1, false, w1op[1],
                                                   (short)0, c1, false, false);

        // bias + ReLU, spill H1 to LDS (C layout -> row-major, pitch 34,
        // bank-conflict-free).
        #pragma unroll
        for (int v = 0; v < 8; ++v) {
            int row = v + 8 * hi;                 // C/D layout: VGPR v, half hi
            h1[row * H1_PITCH + lane15]      = fmaxf(c0[v] + bias1[0], 0.0f);
            h1[row * H1_PITCH + 16 + lane15] = fmaxf(c1[v] + bias1[1], 0.0f);
        }
        // Same-wave DS ops are in-order; wait so the cross-lane reads below
        // see the stores (lanes are lockstep within a wave32).
        asm volatile("s_wait_dscnt 0x0" ::: "memory");

        // ---- Layer 2: [16x32] @ [32x32], 8 K-chunks x 2 N-tiles = 16 WMMAs
        v8f d0 = {};
        v8f d1 = {};
        #pragma unroll
        for (int c = 0; c < 8; ++c) {
            const int kb = 4 * c + 2 * hi;
            const v2f a2 = *(const v2f*)(h1 + lane15 * H1_PITCH + kb);
            d0 = __builtin_amdgcn_wmma_f32_16x16x4_f32(false, a2, false, w2op[c][0],
                                                       (short)0, d0, false, false);
            d1 = __builtin_amdgcn_wmma_f32_16x16x4_f32(false, a2, false, w2op[c][1],
                                                       (short)0, d1, false, false);
        }

        // ---- Layer 3 (30->1): per-lane partials + 16-lane butterfly.
        // Lane holds column n; after xor{1,2,4,8} each 16-lane half has the
        // full dot product: lanes 0-15 carry rows v, lanes 16-31 rows v+8.
        #pragma unroll
        for (int v = 0; v < 8; ++v) {
            float p = fmaxf(d0[v] + bias2[0], 0.0f) * w3a
                    + fmaxf(d1[v] + bias2[1], 0.0f) * w3b;
            p += __shfl_xor(p, 1);
            p += __shfl_xor(p, 2);
            p += __shfl_xor(p, 4);
            p += __shfl_xor(p, 8);
            if (lane15 == 0) td[v + 8 * hi] = sPd - wc * (p + bias3);
        }
        asm volatile("s_wait_dscnt 0x0" ::: "memory");

        // ---- Dispatch + store: 16 samples x 100 units = 1600 contiguous
        //      floats. 12 x B128 + 1 x B64, fully coalesced.
        float* outp = out + (size_t)b0 * NU;
        #pragma unroll 4
        for (int it = 0; it < 12; ++it) {
            const int e = it * 128 + lane * 4;
            v4f r;
            #pragma unroll
            for (int q = 0; q < 4; ++q) {
                int ei = e + q;
                int bl = ei / NU;
                int u  = ei - bl * NU;
                float val = td[bl] - s_cum[u];
                r[q] = fminf(fmaxf(val, 0.0f), s_pmx[u]);
            }
            *(v4f*)(outp + e) = r;
        }
        {
            const int e = 1536 + lane * 2;
            v2f r;
            #pragma unroll
            for (int q = 0; q < 2; ++q) {
                int ei = e + q;
                int bl = ei / NU;
                int u  = ei - bl * NU;
                float val = td[bl] - s_cum[u];
                r[q] = fminf(fmaxf(val, 0.0f), s_pmx[u]);
            }
            *(v2f*)(outp + e) = r;
        }
    }
}

// ---------------------------------------------------------------------------
extern "C" void kernel_launch(void* const* d_in, const int* in_sizes, int n_in,
                              void* d_out, int out_size, void* d_ws, size_t ws_size,
                              hipStream_t stream) {
    const float* x    = (const float*)d_in[0];
    const float* Cost = (const float*)d_in[1];
    const float* Pmax = (const float*)d_in[2];
    const float* Pd   = (const float*)d_in[3];
    const float* wcap = (const float*)d_in[4];
    const float* W1   = (const float*)d_in[5];
    const float* b1   = (const float*)d_in[6];
    const float* W2   = (const float*)d_in[7];
    const float* b2   = (const float*)d_in[8];
    const float* W3   = (const float*)d_in[9];
    const float* b3   = (const float*)d_in[10];
    float* out = (float*)d_out;

    float* ws     = (float*)d_ws;
    float* cum    = ws;          // 100 floats (padded region to 128)
    float* pmx    = ws + 128;    // 100 floats
    float* sumPd  = ws + 256;    // 1 float

    const int B = in_sizes[0] / INDIM;

    setup_units_kernel<<<1, 128, 0, stream>>>(Cost, Pmax, Pd, cum, pmx, sumPd);

    const int samplesPerBlock = 16 * TILES_PER_WAVE * WAVES_PER_BLOCK;  // 512
    const int nBlocks = (B + samplesPerBlock - 1) / samplesPerBlock;    // 512 for B=262144
    mlp_dispatch_kernel<<<nBlocks, WAVES_PER_BLOCK * 32, 0, stream>>>(
        x, W1, b1, W2, b2, W3, b3, wcap, cum, pmx, sumPd, out, B);
}